// AdaptiveGraphAttention_77369540870704
// MI455X (gfx1250) — compile-verified
//
#include <hip/hip_runtime.h>
#include <hip/hip_bf16.h>
#include <math.h>

// Problem constants (from reference)
#define BB   4
#define SS   128
#define DD   256
#define HH   8
#define HDD  32
#define NSS  129
#define NEGV -1000000000.0f

typedef float v2f __attribute__((ext_vector_type(2)));
typedef float v8f __attribute__((ext_vector_type(8)));
typedef unsigned int u32x4 __attribute__((ext_vector_type(4)));
typedef int i32x4 __attribute__((ext_vector_type(4)));
typedef int i32x8 __attribute__((ext_vector_type(8)));

__device__ __forceinline__ float wave_sum(float v) {
    #pragma unroll
    for (int o = 16; o > 0; o >>= 1) v += __shfl_xor(v, o, 32);
    return v;
}
__device__ __forceinline__ float wave_max(float v) {
    #pragma unroll
    for (int o = 16; o > 0; o >>= 1) v = fmaxf(v, __shfl_xor(v, o, 32));
    return v;
}

// ---------------------------------------------------------------------------
// TDM: DMA a 2D tile (tile_d0 x tile_d1, 4-byte elems, row stride = stride_d0
// elements) from global memory into LDS at lds_off. D# layout per CDNA5 ISA
// ch.8 (group0: count/lds/global/type; group1: dims/tiles/strides).
// ---------------------------------------------------------------------------
__device__ __forceinline__ void tdm_load_2d_to_lds(const void* gaddr, unsigned lds_off,
                                                   unsigned tensor_d0, unsigned tensor_d1,
                                                   unsigned tile_d0, unsigned tile_d1,
                                                   unsigned stride_d0)
{
    unsigned long long ga = (unsigned long long)gaddr;
    u32x4 g0;
    g0.x = 1u;                                            // count=1, user descriptor
    g0.y = lds_off;                                       // lds_addr (bytes)
    g0.z = (unsigned)(ga & 0xFFFFFFFFu);                  // global_addr[31:0]
    g0.w = (unsigned)((ga >> 32) & 0x1FFFFFFu) | (2u << 30); // addr[56:32], type=2
    i32x8 g1;
    g1[0] = (int)(2u << 16);                              // data_size=2 (4B), no multicast
    g1[1] = (int)((tensor_d0 & 0xFFFFu) << 16);           // tensor_dim0[15:0] @ bits 63:48
    g1[2] = (int)(((tensor_d0 >> 16) & 0xFFFFu) | ((tensor_d1 & 0xFFFFu) << 16));
    g1[3] = (int)(((tensor_d1 >> 16) & 0xFFFFu) | ((tile_d0 & 0xFFFFu) << 16));
    g1[4] = (int)(tile_d1 & 0xFFFFu);                     // tile_dim1, tile_dim2=0
    g1[5] = (int)stride_d0;                               // tensor_dim0_stride[31:0]
    g1[6] = 0;                                            // stride hi / dim1_stride lo
    g1[7] = 0;                                            // dim1_stride hi
    i32x4 z4 = {0, 0, 0, 0};
#if __clang_major__ >= 23
    i32x8 z8 = {0, 0, 0, 0, 0, 0, 0, 0};
    __builtin_amdgcn_tensor_load_to_lds(g0, g1, z4, z4, z8, 0);
#else
    __builtin_amdgcn_tensor_load_to_lds(g0, g1, z4, z4, 0);
#endif
}

// ---------------------------------------------------------------------------
// Cooperative f32 WMMA GEMM: C[M,N] = act(A'[M,K] @ W[K,N] + bias).
// One workgroup (8 waves) per 16-wide N column; the K x 16 weight slab is
// staged into LDS once via the Tensor Data Mover, then each wave computes one
// 16x16 tile per 8 M-tiles with V_WMMA_F32_16X16X4_F32 (A from global/L2,
// B from LDS, bank-conflict-free).
// A row remap: ar = m + (m/rowDiv)*rowSkip + rowOff  (handles nv[:,1:] views)
// ---------------------------------------------------------------------------
__global__ void gemm_wmma_coop_kernel(const float* __restrict__ A,
                                      const float* __restrict__ W,
                                      const float* __restrict__ bias,
                                      float* __restrict__ C,
                                      int M, int K, int N,
                                      int rowDiv, int rowSkip, int rowOff,
                                      int doRelu)
{
    __shared__ float ldsW[256 * 16];                      // K x 16 slab (16 KB)
    const int lane = threadIdx.x & 31;
    const int wave = threadIdx.x >> 5;
    const int lh   = lane & 15;
    const int kg   = (lane >> 4) << 1;                    // lanes 16-31 hold K+2,K+3
    const int n0   = blockIdx.y << 4;
    const int m0   = (blockIdx.x * 8 + wave) << 4;

    if (wave == 0) {
        const unsigned lds_off = (unsigned)(unsigned long long)(void*)ldsW;
        tdm_load_2d_to_lds(W + n0, lds_off,
                           (unsigned)(N - n0), (unsigned)K,   // tensor dims (OOB-safe)
                           16u, (unsigned)K,                  // tile: 16 cols x K rows
                           (unsigned)N);                      // row stride (elements)
        __builtin_amdgcn_s_wait_tensorcnt(0);
    }
    __syncthreads();

    int mrow = m0 + lh; if (mrow >= M) mrow = M - 1;
    const int ar = mrow + (mrow / rowDiv) * rowSkip + rowOff;
    const float* Ap = A + (size_t)ar * K + kg;
    const float* Bp = ldsW + (size_t)kg * 16 + lh;

    v8f acc = {};
    for (int k = 0; k < K; k += 4) {
        v2f a, b;
        a.x = Ap[k];            a.y = Ap[k + 1];
        b.x = Bp[k * 16];       b.y = Bp[(k + 1) * 16];
        acc = __builtin_amdgcn_wmma_f32_16x16x4_f32(false, a, false, b,
                                                    (short)0, acc, false, false);
    }

    const int n = n0 + lh;
    const float bv = bias ? bias[n] : 0.0f;
    #pragma unroll
    for (int r = 0; r < 8; ++r) {
        const int m = m0 + r + ((lane >> 4) << 3);
        if (m < M) {
            float v = acc[r] + bv;
            if (doRelu) v = fmaxf(v, 0.0f);
            C[(size_t)m * N + n] = v;
        }
    }
}

// ---------------------------------------------------------------------------
// Fused gate: G[b,i,j] = (i==j) ? NEG : sigmoid(dh_i.dt_j + topo) * (vh_i.vt_j)
// Two WMMA accumulators per tile, both in NT form (B loaded transposed).
// ---------------------------------------------------------------------------
__global__ void graph_gate_kernel(const float* __restrict__ dh, const float* __restrict__ dt,
                                  const float* __restrict__ vh, const float* __restrict__ vt,
                                  const float* __restrict__ topo, float* __restrict__ G)
{
    const int lane = threadIdx.x & 31;
    const int lh = lane & 15, kg = (lane >> 4) << 1;
    const int b  = blockIdx.z;
    const int m0 = blockIdx.x << 4, n0 = blockIdx.y << 4;

    const float* dhp = dh + ((size_t)b * SS + m0 + lh) * DD + kg;
    const float* dtp = dt + ((size_t)b * SS + n0 + lh) * DD + kg;
    const float* vhp = vh + ((size_t)b * SS + m0 + lh) * DD + kg;
    const float* vtp = vt + ((size_t)b * SS + n0 + lh) * DD + kg;

    v8f acc1 = {}, acc2 = {};
    for (int k = 0; k < DD; k += 4) {
        v2f a, bm, c, d;
        a.x  = dhp[k]; a.y  = dhp[k + 1];
        bm.x = dtp[k]; bm.y = dtp[k + 1];
        acc1 = __builtin_amdgcn_wmma_f32_16x16x4_f32(false, a, false, bm,
                                                     (short)0, acc1, false, false);
        c.x  = vhp[k]; c.y  = vhp[k + 1];
        d.x  = vtp[k]; d.y  = vtp[k + 1];
        acc2 = __builtin_amdgcn_wmma_f32_16x16x4_f32(false, c, false, d,
                                                     (short)0, acc2, false, false);
    }
    const float tb = topo[0];
    #pragma unroll
    for (int r = 0; r < 8; ++r) {
        const int i = m0 + r + ((lane >> 4) << 3);
        const int j = n0 + lh;
        float g;
        if (i == j) g = NEGV;
        else {
            float s = 1.0f / (1.0f + expf(-(acc1[r] + tb)));
            g = s * acc2[r];
        }
        G[((size_t)b * SS + i) * SS + j] = g;
    }
}

// L2-normalize rows of X (rows x 256), one wave per row.
__global__ void rownorm_kernel(float* __restrict__ X, int rows)
{
    const int row = blockIdx.x * (blockDim.x >> 5) + (threadIdx.x >> 5);
    if (row >= rows) return;
    const int lane = threadIdx.x & 31;
    float* x = X + (size_t)row * DD;
    float v[8], s = 0.0f;
    #pragma unroll
    for (int c = 0; c < 8; ++c) { v[c] = x[lane + 32 * c]; s += v[c] * v[c]; }
    s = wave_sum(s);
    const float inv = 1.0f / sqrtf(s);
    #pragma unroll
    for (int c = 0; c < 8; ++c) x[lane + 32 * c] = v[c] * inv;
}

// Threshold-masked, renormalized softmax per row of G (128 wide).
__global__ void adj_kernel(const float* __restrict__ G, const float* __restrict__ alpha_p,
                           float* __restrict__ adj)
{
    const int row = blockIdx.x * (blockDim.x >> 5) + (threadIdx.x >> 5);
    if (row >= BB * SS) return;
    const int lane = threadIdx.x & 31;
    const float* g = G + (size_t)row * SS;
    float v[4], m = -3.4e38f;
    #pragma unroll
    for (int t = 0; t < 4; ++t) { v[t] = g[lane + 32 * t]; m = fmaxf(m, v[t]); }
    m = wave_max(m);
    const float alpha = fminf(fmaxf(alpha_p[0], 1e-5f), 1.0f);
    const float thr = m * alpha;
    float s = 0.0f;
    #pragma unroll
    for (int t = 0; t < 4; ++t) {
        v[t] = (v[t] < thr) ? 0.0f : expf(v[t] - m);
        s += v[t];
    }
    s = wave_sum(s);
    const float inv = (s > 0.0f) ? 1.0f / s : 0.0f;
    float* a = adj + (size_t)row * SS;
    #pragma unroll
    for (int t = 0; t < 4; ++t) a[lane + 32 * t] = v[t] * inv;
}

// Fold eu_w2 with per-head we: w2e[k][h] = sum_d eu_w2[k, h*32+d]*we[d]
__global__ void fold_w2_kernel(const float* __restrict__ eu_w2, const float* __restrict__ eu_b2,
                               const float* __restrict__ ap_w,
                               float* __restrict__ w2e, float* __restrict__ b2e)
{
    const int t = blockIdx.x * blockDim.x + threadIdx.x;
    const float* we = ap_w + 2 * HDD;
    if (t < DD * HH) {
        const int k = t >> 3, h = t & 7;
        float s = 0.0f;
        for (int d = 0; d < HDD; ++d) s += eu_w2[(size_t)k * DD + h * HDD + d] * we[d];
        w2e[t] = s;
    }
    if (t < HH) {
        float s = 0.0f;
        for (int d = 0; d < HDD; ++d) s += eu_b2[t * HDD + d] * we[d];
        b2e[t] = s;
    }
}

// Per-(b,h,i): qdot = q_head . wq ; kdot = k_head . wk
__global__ void qk_dot_kernel(const float* __restrict__ q, const float* __restrict__ k,
                              const float* __restrict__ ap_w,
                              float* __restrict__ qdot, float* __restrict__ kdot)
{
    const int t = blockIdx.x * blockDim.x + threadIdx.x;
    if (t >= BB * HH * NSS) return;
    const int b = t / (HH * NSS), rem = t % (HH * NSS);
    const int h = rem / NSS, i = rem % NSS;
    const float* qr = q + ((size_t)b * NSS + i) * DD + h * HDD;
    const float* kr = k + ((size_t)b * NSS + i) * DD + h * HDD;
    float sq = 0.0f, sk = 0.0f;
    for (int d = 0; d < HDD; ++d) { sq += qr[d] * ap_w[d]; sk += kr[d] * ap_w[HDD + d]; }
    qdot[t] = sq; kdot[t] = sk;
}

// Per-edge: e1 = P_i' + Q_j' + b1; LayerNorm; relu; dot with folded w2e;
// sE[b,h,i,j] = new_adj==0 ? NEG : new_adj * (se_h + b2e_h)
__global__ void edge_score_kernel(const float* __restrict__ P, const float* __restrict__ Q,
                                  const float* __restrict__ eu_b1,
                                  const float* __restrict__ eu_g, const float* __restrict__ eu_beta,
                                  const float* __restrict__ w2e, const float* __restrict__ b2e,
                                  const float* __restrict__ adj, float* __restrict__ sE)
{
    const int e = blockIdx.x * (blockDim.x >> 5) + (threadIdx.x >> 5);
    if (e >= BB * NSS * NSS) return;
    const int lane = threadIdx.x & 31;
    const int b = e / (NSS * NSS), rem = e % (NSS * NSS);
    const int i = rem / NSS, j = rem % NSS;

    const float* Pr = nullptr;
    const float* Qr = nullptr;
    if (i >= 1)      Pr = P + ((size_t)b * SS + (i - 1)) * DD;
    else if (j >= 1) Pr = P + ((size_t)b * SS + (j - 1)) * DD;
    if (j >= 1)      Qr = Q + ((size_t)b * SS + (j - 1)) * DD;
    else if (i >= 1) Qr = Q + ((size_t)b * SS + (i - 1)) * DD;

    float x[8], s = 0.0f;
    #pragma unroll
    for (int c = 0; c < 8; ++c) {
        const int d = lane + 32 * c;
        float v = eu_b1[d];
        if (Pr) v += Pr[d];
        if (Qr) v += Qr[d];
        x[c] = v; s += v;
    }
    s = wave_sum(s);
    const float mean = s * (1.0f / DD);
    float vs = 0.0f;
    #pragma unroll
    for (int c = 0; c < 8; ++c) { float t = x[c] - mean; vs += t * t; }
    vs = wave_sum(vs);
    const float inv = 1.0f / sqrtf(vs * (1.0f / DD) + 1e-5f);

    float acc[8] = {0, 0, 0, 0, 0, 0, 0, 0};
    #pragma unroll
    for (int c = 0; c < 8; ++c) {
        const int d = lane + 32 * c;
        const float r = fmaxf((x[c] - mean) * inv * eu_g[d] + eu_beta[d], 0.0f);
        #pragma unroll
        for (int h = 0; h < HH; ++h) acc[h] += r * w2e[d * HH + h];
    }
    #pragma unroll
    for (int h = 0; h < HH; ++h) acc[h] = wave_sum(acc[h]);

    float na;
    if (i == 0 && j >= 1)      na = 1.0f;
    else if (i >= 1 && j >= 1) na = adj[((size_t)b * SS + (i - 1)) * SS + (j - 1)];
    else                       na = 0.0f;

    if (lane < HH) {
        const float se = acc[lane] + b2e[lane];
        const float val = (na == 0.0f) ? NEGV : na * se;
        sE[(((size_t)b * HH + lane) * NSS + i) * NSS + j] = val;
    }
}

// Per-(b,h,i) softmax over j of qdot_i + kdot_j + ap_b + sE ; writes attn (d_out)
__global__ void attn_softmax_kernel(const float* __restrict__ sE, const float* __restrict__ qdot,
                                    const float* __restrict__ kdot, const float* __restrict__ ap_b,
                                    float* __restrict__ attn)
{
    const int row = blockIdx.x * (blockDim.x >> 5) + (threadIdx.x >> 5);
    if (row >= BB * HH * NSS) return;
    const int lane = threadIdx.x & 31;
    const int b = row / (HH * NSS), rem = row % (HH * NSS);
    const int h = rem / NSS;
    const float qd = qdot[row] + ap_b[0];
    const float* se = sE + (size_t)row * NSS;
    const float* kd = kdot + ((size_t)b * HH + h) * NSS;

    float v[5], m = -3.4e38f;
    #pragma unroll
    for (int t = 0; t < 5; ++t) {
        const int j = lane + 32 * t;
        if (j < NSS) { v[t] = qd + kd[j] + se[j]; m = fmaxf(m, v[t]); }
        else v[t] = -3.4e38f;
    }
    m = wave_max(m);
    float s = 0.0f;
    #pragma unroll
    for (int t = 0; t < 5; ++t) {
        const int j = lane + 32 * t;
        if (j < NSS) { v[t] = expf(v[t] - m); s += v[t]; }
    }
    s = wave_sum(s);
    const float inv = 1.0f / s;
    #pragma unroll
    for (int t = 0; t < 5; ++t) {
        const int j = lane + 32 * t;
        if (j < NSS) attn[(size_t)row * NSS + j] = v[t] * inv;
    }
}

// ctx[b,i,h*32+n] = sum_j attn[b,h,i,j] * v[b,j,h*32+n]  (WMMA, K padded 129->132)
__global__ void ctx_wmma_kernel(const float* __restrict__ attn, const float* __restrict__ v,
                                float* __restrict__ ctx)
{
    const int lane = threadIdx.x & 31;
    const int lh = lane & 15, kg = (lane >> 4) << 1;
    const int bh = blockIdx.z, b = bh >> 3, h = bh & 7;
    const int m0 = blockIdx.x << 4, n0 = blockIdx.y << 4;
    const float* At = attn + (size_t)bh * NSS * NSS;
    const float* Vb = v + (size_t)b * NSS * DD + h * HDD;
    const int i = m0 + lh;

    v8f acc = {};
    for (int k = 0; k < 132; k += 4) {
        const int k0 = k + kg, k1 = k + kg + 1;
        v2f a, bm;
        a.x  = (i < NSS && k0 < NSS) ? At[(size_t)i * NSS + k0] : 0.0f;
        a.y  = (i < NSS && k1 < NSS) ? At[(size_t)i * NSS + k1] : 0.0f;
        bm.x = (k0 < NSS) ? Vb[(size_t)k0 * DD + n0 + lh] : 0.0f;
        bm.y = (k1 < NSS) ? Vb[(size_t)k1 * DD + n0 + lh] : 0.0f;
        acc = __builtin_amdgcn_wmma_f32_16x16x4_f32(false, a, false, bm,
                                                    (short)0, acc, false, false);
    }
    #pragma unroll
    for (int r = 0; r < 8; ++r) {
        const int m = m0 + r + ((lane >> 4) << 3);
        if (m < NSS) ctx[((size_t)b * NSS + m) * DD + h * HDD + n0 + lh] = acc[r];
    }
}

// ---------------------------------------------------------------------------
extern "C" void kernel_launch(void* const* d_in, const int* in_sizes, int n_in,
                              void* d_out, int out_size, void* d_ws, size_t ws_size,
                              hipStream_t stream) {
    const float* desc    = (const float*)d_in[0];   // (4,128,256)
    const float* nv      = (const float*)d_in[1];   // (4,129,256)
    const float* th_w1   = (const float*)d_in[2];
    const float* th_b1   = (const float*)d_in[3];
    const float* th_w2   = (const float*)d_in[4];
    const float* th_b2   = (const float*)d_in[5];
    const float* tt_w1   = (const float*)d_in[6];
    const float* tt_b1   = (const float*)d_in[7];
    const float* tt_w2   = (const float*)d_in[8];
    const float* tt_b2   = (const float*)d_in[9];
    const float* ah_w    = (const float*)d_in[10];
    const float* ah_b    = (const float*)d_in[11];
    const float* at_w    = (const float*)d_in[12];
    const float* at_b    = (const float*)d_in[13];
    const float* q_w     = (const float*)d_in[14];
    const float* q_b     = (const float*)d_in[15];
    const float* k_w     = (const float*)d_in[16];
    const float* k_b     = (const float*)d_in[17];
    const float* v_w     = (const float*)d_in[18];
    const float* v_b     = (const float*)d_in[19];
    const float* eu_w1   = (const float*)d_in[20];  // (512,256)
    const float* eu_b1   = (const float*)d_in[21];
    const float* eu_g    = (const float*)d_in[22];
    const float* eu_beta = (const float*)d_in[23];
    const float* eu_w2   = (const float*)d_in[24];
    const float* eu_b2   = (const float*)d_in[25];
    const float* ap_w    = (const float*)d_in[26];  // 96
    const float* ap_b    = (const float*)d_in[27];
    const float* out_w   = (const float*)d_in[28];
    const float* out_b   = (const float*)d_in[29];
    const float* topo    = (const float*)d_in[30];
    const float* alpha   = (const float*)d_in[31];

    float* out      = (float*)d_out;
    float* out_ctx  = out;                          // 4*129*256 = 132096
    float* out_attn = out + (size_t)BB * NSS * DD;  // 4*8*129*129 = 532512

    // Workspace carve-up (floats)
    float* ws = (float*)d_ws;
    size_t o = 0;
    float* dh   = ws + o; o += (size_t)BB * SS * DD;   // 131072
    float* dt   = ws + o; o += (size_t)BB * SS * DD;
    float* tmp  = ws + o; o += (size_t)BB * SS * DD;
    float* vh   = ws + o; o += (size_t)BB * SS * DD;
    float* vt   = ws + o; o += (size_t)BB * SS * DD;
    float* G    = ws + o; o += (size_t)BB * SS * SS;   // 65536
    float* adjb = ws + o; o += (size_t)BB * SS * SS;
    float* P    = ws + o; o += (size_t)BB * SS * DD;
    float* Q    = ws + o; o += (size_t)BB * SS * DD;
    float* qm   = ws + o; o += (size_t)BB * NSS * DD;  // 132096
    float* km   = ws + o; o += (size_t)BB * NSS * DD;
    float* vm   = ws + o; o += (size_t)BB * NSS * DD;
    float* ctx  = ws + o; o += (size_t)BB * NSS * DD;
    float* w2e  = ws + o; o += (size_t)DD * HH;        // 2048
    float* b2e  = ws + o; o += (size_t)HH;
    float* qdot = ws + o; o += (size_t)BB * HH * NSS;  // 4128
    float* kdot = ws + o; o += (size_t)BB * HH * NSS;
    float* sE   = ws + o; o += (size_t)BB * HH * NSS * NSS; // 532512
    (void)ws_size; (void)n_in; (void)in_sizes; (void)out_size;

    const int BIG = 1 << 30;
    // Cooperative GEMM: 8 waves/block, 8 M-tiles/block, one N column per block.
    const dim3 g512(4, 16), g516(5, 16);

    // Topology-head / tail MLPs: dh, dt
    gemm_wmma_coop_kernel<<<g512, 256, 0, stream>>>(desc, th_w1, th_b1, tmp, 512, 256, 256, BIG, 0, 0, 1);
    gemm_wmma_coop_kernel<<<g512, 256, 0, stream>>>(tmp, th_w2, th_b2, dh, 512, 256, 256, BIG, 0, 0, 0);
    gemm_wmma_coop_kernel<<<g512, 256, 0, stream>>>(desc, tt_w1, tt_b1, tmp, 512, 256, 256, BIG, 0, 0, 1);
    gemm_wmma_coop_kernel<<<g512, 256, 0, stream>>>(tmp, tt_w2, tt_b2, dt, 512, 256, 256, BIG, 0, 0, 0);
    rownorm_kernel<<<256, 128, 0, stream>>>(dh, 2 * BB * SS);  // dh,dt contiguous

    // vh, vt from nv[:,1:] (row remap m -> m + m/128 + 1)
    gemm_wmma_coop_kernel<<<g512, 256, 0, stream>>>(nv, ah_w, ah_b, vh, 512, 256, 256, SS, 1, 1, 0);
    gemm_wmma_coop_kernel<<<g512, 256, 0, stream>>>(nv, at_w, at_b, vt, 512, 256, 256, SS, 1, 1, 0);

    // G = sigmoid(dh.dtT + topo) * (vh.vtT), diag = NEG; then adj
    graph_gate_kernel<<<dim3(8, 8, BB), 32, 0, stream>>>(dh, dt, vh, vt, topo, G);
    adj_kernel<<<128, 128, 0, stream>>>(G, alpha, adjb);

    // Factored edge first layer: P = desc@W1a, Q = desc@W1b
    gemm_wmma_coop_kernel<<<g512, 256, 0, stream>>>(desc, eu_w1, nullptr, P, 512, 256, 256, BIG, 0, 0, 0);
    gemm_wmma_coop_kernel<<<g512, 256, 0, stream>>>(desc, eu_w1 + 256 * 256, nullptr, Q, 512, 256, 256, BIG, 0, 0, 0);
    fold_w2_kernel<<<8, 256, 0, stream>>>(eu_w2, eu_b2, ap_w, w2e, b2e);

    // q, k, v projections on full nv (M = 516)
    gemm_wmma_coop_kernel<<<g516, 256, 0, stream>>>(nv, q_w, q_b, qm, 516, 256, 256, BIG, 0, 0, 0);
    gemm_wmma_coop_kernel<<<g516, 256, 0, stream>>>(nv, k_w, k_b, km, 516, 256, 256, BIG, 0, 0, 0);
    gemm_wmma_coop_kernel<<<g516, 256, 0, stream>>>(nv, v_w, v_b, vm, 516, 256, 256, BIG, 0, 0, 0);
    qk_dot_kernel<<<17, 256, 0, stream>>>(qm, km, ap_w, qdot, kdot);

    // Edge scores (LN + relu + folded 256x8 matvec), one wave per edge
    const int nEdges = BB * NSS * NSS;  // 66564
    edge_score_kernel<<<(nEdges + 7) / 8, 256, 0, stream>>>(P, Q, eu_b1, eu_g, eu_beta,
                                                            w2e, b2e, adjb, sE);

    // Softmax -> attn (second output), then ctx = attn @ v, then out projection
    attn_softmax_kernel<<<(BB * HH * NSS + 3) / 4, 128, 0, stream>>>(sE, qdot, kdot, ap_b, out_attn);
    ctx_wmma_kernel<<<dim3(9, 2, BB * HH), 32, 0, stream>>>(out_attn, vm, ctx);
    gemm_wmma_coop_kernel<<<g516, 256, 0, stream>>>(ctx, out_w, out_b, out_ctx, 516, 256, 256, BIG, 0, 0, 0);
}